// AttentionDecoderRNN_23579370455621
// MI455X (gfx1250) — compile-verified
//
#include <hip/hip_runtime.h>
#include <math.h>

#define H 1024
#define V 50257
#define L 512

typedef __attribute__((ext_vector_type(2))) float v2f;
typedef __attribute__((ext_vector_type(4))) float v4f;
typedef __attribute__((ext_vector_type(8))) float v8f;

// ---------------------------------------------------------------------------
// GEMV  y[r] = act( sum_k W[r,k]*x[k] + b[r] )  via V_WMMA_F32_16X16X4_F32.
//
// One wave per 16-row tile. A-operand = x replicated across the 16 rows
// (documented f32 16x4 A layout: vgpr0 = K0|K2 by lane-half, vgpr1 = K1|K3),
// served from LDS. B-operand = the 16 weight rows as columns (layout
// symmetric with A, consistent with the ISA §7.12.4 B pattern: low lanes
// hold the first half of K). Weight stream uses B128 loads: lanes 0-15 fetch
// row bytes [k,k+4), lanes 16-31 fetch [k+4,k+8); 4 shfl_xor(16) ops
// redistribute the halves into the two WMMA B operands. Every row of D holds
// the same 16 outputs; row 0 = C-vgpr0, lanes 0..15. K % 8 == 0 required.
//
// NT=true marks the weight stream non-temporal so the 206 MB out_W pass
// doesn't evict the ~38 MB of small weights from MI455X's 192 MB L2.
// ---------------------------------------------------------------------------
template <int ACT, bool NT>
__global__ void gemv_wmma_k(const float* __restrict__ W, const float* __restrict__ x,
                            const float* __restrict__ bias, float* __restrict__ y,
                            int rows, int K) {
    __shared__ float xs[2048];                         // K <= 2048
    for (int i = threadIdx.x; i < K; i += blockDim.x) xs[i] = x[i];
    __syncthreads();

    const int lane = threadIdx.x & 31;
    const int wave = (int)((blockIdx.x * blockDim.x + threadIdx.x) >> 5);
    const int r0   = wave * 16;
    if (r0 >= rows) return;                 // wave-uniform: EXEC stays all-ones

    const int m    = lane & 15;
    int r = r0 + m;
    if (r >= rows) r = rows - 1;            // per-lane clamp (select, no branch)
    const int half = lane >> 4;

    const float* __restrict__ Wr = W + (size_t)r * K + 4 * half;
    const float* __restrict__ xh = xs + 2 * half;

    v8f c0 = {};
    v8f c1 = {};
    for (int k = 0; k < K; k += 8) {
        v4f f = NT ? __builtin_nontemporal_load((const v4f*)(Wr + k))
                   : *(const v4f*)(Wr + k);
        v2f a0 = { xh[k],     xh[k + 1] };             // LDS (DS pipe)
        v2f a1 = { xh[k + 4], xh[k + 5] };
        // exchange K-halves between lane L and lane L+16
        float sx = __shfl_xor(f.x, 16);
        float sy = __shfl_xor(f.y, 16);
        float sz = __shfl_xor(f.z, 16);
        float sw = __shfl_xor(f.w, 16);
        v2f b0, b1;
        b0.x = half ? sz : f.x;  b0.y = half ? sw : f.y;   // K0|K2, K1|K3
        b1.x = half ? f.z : sx;  b1.y = half ? f.w : sy;   // K4|K6, K5|K7
        c0 = __builtin_amdgcn_wmma_f32_16x16x4_f32(false, a0, false, b0,
                                                   (short)0, c0, false, false);
        c1 = __builtin_amdgcn_wmma_f32_16x16x4_f32(false, a1, false, b1,
                                                   (short)0, c1, false, false);
    }
    float v = c0[0] + c1[0];                // D row 0: vgpr0, lanes 0..15
    if (lane < 16) {
        int rr = r0 + lane;
        if (rr < rows) {
            float out = v + bias[rr];
            if (ACT == 1) out = fmaxf(out, 0.0f);
            y[rr] = out;
        }
    }
}

// --------------------------- elementwise / glue ----------------------------
__global__ void prep_k(const int* __restrict__ tok, const float* __restrict__ hidden,
                       const float* __restrict__ emb, float* __restrict__ cat1,
                       float* __restrict__ cat2) {
    int i = blockIdx.x * blockDim.x + threadIdx.x;
    if (i < H) {
        float e = emb[(size_t)tok[0] * H + i];
        cat1[i]     = e;          // [e0, h0]
        cat1[H + i] = hidden[i];
        cat2[i]     = e;          // [e0, attn_applied] (2nd half filled later)
    }
}

__global__ void softmax_L_k(const float* __restrict__ logits,
                            float* __restrict__ w_out, float* __restrict__ w_ws) {
    __shared__ float red[L];
    int i = threadIdx.x;                    // 512 threads
    float v = logits[i];
    red[i] = v; __syncthreads();
    for (int s = L / 2; s > 0; s >>= 1) {
        if (i < s) red[i] = fmaxf(red[i], red[i + s]);
        __syncthreads();
    }
    float mx = red[0]; __syncthreads();
    float e = expf(v - mx);
    red[i] = e; __syncthreads();
    for (int s = L / 2; s > 0; s >>= 1) {
        if (i < s) red[i] += red[i + s];
        __syncthreads();
    }
    float w = e / red[0];
    w_out[i] = w;
    w_ws[i]  = w;
}

__global__ void attn_apply_k(const float* __restrict__ w, const float* __restrict__ enc,
                             float* __restrict__ cat2) {
    int h = blockIdx.x * blockDim.x + threadIdx.x;
    if (h < H) {
        float s = 0.0f;
        for (int l = 0; l < L; ++l)                 // coalesced across h
            s = fmaf(w[l], enc[(size_t)l * H + h], s);
        cat2[H + h] = s;
    }
}

__device__ __forceinline__ float sigmoidf_(float x) { return 1.0f / (1.0f + expf(-x)); }

__global__ void gru_combine_k(const float* __restrict__ gi, const float* __restrict__ gh,
                              const float* __restrict__ h0, float* __restrict__ hn_ws,
                              float* __restrict__ hn_out) {
    int h = blockIdx.x * blockDim.x + threadIdx.x;
    if (h < H) {
        float r  = sigmoidf_(gi[h]         + gh[h]);
        float z  = sigmoidf_(gi[H + h]     + gh[H + h]);
        float n  = tanhf(gi[2 * H + h] + r * gh[2 * H + h]);
        float hv = (1.0f - z) * n + z * h0[h];
        hn_ws[h]  = hv;
        hn_out[h] = hv;
    }
}

__global__ void log_softmax_V_k(const float* __restrict__ logits, float* __restrict__ out) {
    __shared__ float red[1024];
    int t = threadIdx.x;                    // 1024 threads, single block
    float mx = -INFINITY;
    for (int i = t; i < V; i += 1024) mx = fmaxf(mx, logits[i]);
    red[t] = mx; __syncthreads();
    for (int s = 512; s > 0; s >>= 1) {
        if (t < s) red[t] = fmaxf(red[t], red[t + s]);
        __syncthreads();
    }
    mx = red[0]; __syncthreads();
    float sum = 0.0f;
    for (int i = t; i < V; i += 1024) sum += expf(logits[i] - mx);
    red[t] = sum; __syncthreads();
    for (int s = 512; s > 0; s >>= 1) {
        if (t < s) red[t] += red[t + s];
        __syncthreads();
    }
    float logZ = mx + logf(red[0]);
    for (int i = t; i < V; i += 1024) out[i] = logits[i] - logZ;
}

// ---------------------------------------------------------------------------
extern "C" void kernel_launch(void* const* d_in, const int* in_sizes, int n_in,
                              void* d_out, int out_size, void* d_ws, size_t ws_size,
                              hipStream_t stream) {
    (void)in_sizes; (void)n_in; (void)out_size; (void)ws_size;

    const int*   tok    = (const int*)  d_in[0];
    const float* hidden = (const float*)d_in[1];   // (1,1,H)
    const float* enc    = (const float*)d_in[2];   // (L,H)
    const float* emb    = (const float*)d_in[3];   // (V,H)
    const float* attn_W = (const float*)d_in[4];   // (L,2H)
    const float* attn_b = (const float*)d_in[5];
    const float* comb_W = (const float*)d_in[6];   // (H,2H)
    const float* comb_b = (const float*)d_in[7];
    const float* W_ih   = (const float*)d_in[8];   // (3H,H)
    const float* W_hh   = (const float*)d_in[9];   // (3H,H)
    const float* b_ih   = (const float*)d_in[10];
    const float* b_hh   = (const float*)d_in[11];
    const float* out_W  = (const float*)d_in[12];  // (V,H)
    const float* out_b  = (const float*)d_in[13];

    float* out = (float*)d_out;
    float* log_probs = out;                 // [0, V)
    float* hnew_out  = out + V;             // [V, V+H)
    float* attnw_out = out + V + H;         // [V+H, V+H+L)

    // workspace layout (floats)
    float* ws      = (float*)d_ws;
    float* cat1    = ws;                    // 2048
    float* cat2    = ws + 2048;             // 2048
    float* logitsL = ws + 4096;             // 512
    float* attnw   = ws + 4608;             // 512
    float* xvec    = ws + 5120;             // 1024
    float* gi      = ws + 6144;             // 3072
    float* gh      = ws + 9216;             // 3072
    float* hnew    = ws + 12288;            // 1024
    float* logitsV = ws + 13312;            // 50257

    auto gemv_blocks = [](int rows) {       // 8 waves (tiles) per 256-thread block
        int tiles = (rows + 15) / 16;
        return (tiles + 7) / 8;
    };

    prep_k<<<(H + 255) / 256, 256, 0, stream>>>(tok, hidden, emb, cat1, cat2);

    gemv_wmma_k<0, false><<<gemv_blocks(L), 256, 0, stream>>>(attn_W, cat1, attn_b,
                                                              logitsL, L, 2 * H);
    softmax_L_k<<<1, L, 0, stream>>>(logitsL, attnw_out, attnw);

    attn_apply_k<<<(H + 255) / 256, 256, 0, stream>>>(attnw, enc, cat2);

    gemv_wmma_k<1, false><<<gemv_blocks(H), 256, 0, stream>>>(comb_W, cat2, comb_b,
                                                              xvec, H, 2 * H);  // relu

    gemv_wmma_k<0, false><<<gemv_blocks(3 * H), 256, 0, stream>>>(W_ih, xvec, b_ih,
                                                                  gi, 3 * H, H);
    gemv_wmma_k<0, false><<<gemv_blocks(3 * H), 256, 0, stream>>>(W_hh, hidden, b_hh,
                                                                  gh, 3 * H, H);
    gru_combine_k<<<(H + 255) / 256, 256, 0, stream>>>(gi, gh, hidden, hnew, hnew_out);

    // Dominant kernel: 206 MB non-temporal stream (keeps small weights hot in
    // the 192 MB L2 across graph replays). 3142 tiles -> 393 blocks of 8 waves.
    gemv_wmma_k<0, true><<<gemv_blocks(V), 256, 0, stream>>>(out_W, hnew, out_b,
                                                             logitsV, V, H);
    log_softmax_V_k<<<1, 1024, 0, stream>>>(logitsV, log_probs);
}